// BlurLayer_58669253263530
// MI455X (gfx1250) — compile-verified
//
#include <hip/hip_runtime.h>
#include <hip/hip_bf16.h>

// ---------------------------------------------------------------------------
// Motion-blur: per-sample rotated line kernel, depthwise SAME conv.
// B=128, H=W=224, C=3, kernel padded to 32x32, pad_lo=15, pad_hi=16.
// (1) deterministically compact non-zero rotated taps per sample (wave32
//     ballot-rank compaction), (2) TDM (tensor_load_to_lds) stages the
//     63x63x3 halo tile AND the tap list into LDS, then 256 threads do the
//     sparse tap accumulation from LDS. Memory-bound design: ~154 MB HBM
//     traffic -> ~6.6 us floor at 23.3 TB/s; WMMA deliberately unused
//     (depthwise conv does not factor into A[m,k]*B[k,n]).
// ---------------------------------------------------------------------------

#define KK 32
#define IMG 224
#define CH 3
#define TILE 32
#define PAD_LO 15
#define PAD_HI 16
#define HALO (TILE + PAD_LO + PAD_HI)   // 63
#define MAX_TAPS 1024
#define NB 128

typedef unsigned int uint32x4 __attribute__((ext_vector_type(4)));
typedef int int32x8 __attribute__((ext_vector_type(8)));
typedef int int32x4 __attribute__((ext_vector_type(4)));

// ---------------------------------------------------------------------------
// Kernel 1: rotate table kernel per sample, compact non-zero taps in
// deterministic row-major order. One wave32 per sample.
// ---------------------------------------------------------------------------
__global__ __launch_bounds__(32) void prep_taps_kernel(
    const float* __restrict__ table,   // (32, 32, 32, 3)
    const int* __restrict__ amt,       // (B,)
    const int* __restrict__ angles,    // (B,)
    int* __restrict__ counts,          // (B,)
    float2* __restrict__ taps)         // (B, 1024) {w, packed(dy,dx)}
{
    const int b = blockIdx.x;
    const int lane = threadIdx.x;

    const int a = amt[b];
    const float rad = (float)angles[b] * 0.017453292519943295f;
    const float cs = cosf(rad);
    const float sn = sinf(rad);
    const float e = (float)(KK - 1);
    const float x_off = (e - (cs * e - sn * e)) * 0.5f;
    const float y_off = (e - (sn * e + cs * e)) * 0.5f;

    const float* __restrict__ trow = table + (size_t)a * (KK * KK * CH);

    int base = 0;
    #pragma unroll 4
    for (int chunk = 0; chunk < (KK * KK) / 32; ++chunk) {
        const int idx = chunk * 32 + lane;
        const int yy = idx >> 5;   // ky
        const int xx = idx & 31;   // kx
        // reference _rotate_nearest (jnp.round == rintf: half-to-even)
        const float sx = cs * (float)xx - sn * (float)yy + x_off;
        const float sy = sn * (float)xx + cs * (float)yy + y_off;
        const int ix = (int)rintf(sx);
        const int iy = (int)rintf(sy);
        const bool valid = (ix >= 0) && (ix < KK) && (iy >= 0) && (iy < KK);
        float w = 0.0f;
        if (valid) w = trow[(iy * KK + ix) * CH];   // channels identical
        const bool nz = (w != 0.0f);
#if __has_builtin(__builtin_amdgcn_ballot_w32)
        const unsigned mask = __builtin_amdgcn_ballot_w32(nz);
#else
        const unsigned mask = (unsigned)__ballot(nz);
#endif
        if (nz) {
            const int rank = __popc(mask & ((1u << lane) - 1u));
            const int dy = yy - PAD_LO;
            const int dx = xx - PAD_LO;
            const unsigned packed =
                (((unsigned)(dy & 0xFFFF)) << 16) | (unsigned)(dx & 0xFFFF);
            float2 t;
            t.x = w;
            t.y = __int_as_float((int)packed);
            taps[(size_t)b * MAX_TAPS + base + rank] = t;
        }
        base += __popc(mask);
    }
    if (lane == 0) counts[b] = base;
}

// ---------------------------------------------------------------------------
// TDM helper: issue one 2D tensor_load_to_lds. Dimensions in 4-byte elements;
// tensor dims == tile dims (caller pre-clips, so no OOB), row stride may
// exceed the tile width. Caller must s_wait_tensorcnt + barrier.
// ---------------------------------------------------------------------------
#if __has_builtin(__builtin_amdgcn_tensor_load_to_lds)
#define HAVE_TDM 1
__device__ __forceinline__ void tdm_load_2d(unsigned lds_off,
                                            const void* gptr,
                                            unsigned rowElems,
                                            unsigned rows,
                                            unsigned rowStrideElems) {
    const unsigned long long gaddr = (unsigned long long)(size_t)gptr;

    // D# group 0: count=1 (valid), lds_addr, global_addr(57b), type=2
    uint32x4 g0;
    g0[0] = 1u;
    g0[1] = lds_off;
    g0[2] = (unsigned)(gaddr & 0xFFFFFFFFull);
    g0[3] = (unsigned)((gaddr >> 32) & 0x01FFFFFFull) | 0x80000000u;

    // D# group 1: data_size=4B (code 2), tensor_dim0/1, tile_dim0/1, stride0.
    int32x8 g1;
    g1[0] = (int)(2u << 16);                                        // 4B elems
    g1[1] = (int)((rowElems & 0xFFFFu) << 16);                      // tdim0 lo
    g1[2] = (int)((rowElems >> 16) | ((rows & 0xFFFFu) << 16));     // hi | tdim1 lo
    g1[3] = (int)((rows >> 16) | (rowElems << 16));                 // hi | tile0
    g1[4] = (int)rows;                                              // tile1 (tile2=0)
    g1[5] = (int)rowStrideElems;                                    // stride0 lo32
    g1[6] = 0;                                                      // stride0 hi16
    g1[7] = 0;

    int32x4 gz4 = {0, 0, 0, 0};            // 2D: groups 2/3 unused
    int32x8 gz8 = {0, 0, 0, 0, 0, 0, 0, 0};
    __builtin_amdgcn_tensor_load_to_lds(g0, g1, gz4, gz4, gz8, 0);
}
#else
#define HAVE_TDM 0
#endif

// ---------------------------------------------------------------------------
// Kernel 2: sparse depthwise conv. One block per 32x32 output tile per sample.
// Halo tile (<=63x63x3 f32, 47.6 KB) + tap list staged to LDS via TDM.
// ---------------------------------------------------------------------------
__global__ __launch_bounds__(256) void blur_conv_kernel(
    const float* __restrict__ x,       // (B, 224, 224, 3)
    const int* __restrict__ counts,
    const float2* __restrict__ taps,
    float* __restrict__ out)
{
    __shared__ float tile[HALO * HALO * CH];   // packed [ycount][xcount*3]
    __shared__ float2 staps[MAX_TAPS];

    const int b = blockIdx.z;
    const int tx0 = blockIdx.x * TILE;
    const int ty0 = blockIdx.y * TILE;
    const int tid = threadIdx.x;

    // Clipped halo extent (pixels outside the image are SAME zero-padding).
    const int y0 = max(ty0 - PAD_LO, 0);
    const int y1 = min(ty0 + TILE - 1 + PAD_HI, IMG - 1);
    const int x0 = max(tx0 - PAD_LO, 0);
    const int x1 = min(tx0 + TILE - 1 + PAD_HI, IMG - 1);
    const int ycount = y1 - y0 + 1;
    const int xcount = x1 - x0 + 1;
    const int rowElems = xcount * CH;

    const float* __restrict__ src =
        x + (((size_t)b * IMG + y0) * IMG + x0) * CH;
    const int count = counts[b];

#if HAVE_TDM
    if (tid < 32) {   // wave-uniform branch: only wave 0 issues TDM ops
        // (a) halo image tile: ycount rows of rowElems f32, stride 224*3
        tdm_load_2d((unsigned)(size_t)(void*)&tile[0], src,
                    (unsigned)rowElems, (unsigned)ycount,
                    (unsigned)(IMG * CH));
        // (b) tap list: 1 row of 2*count f32 (float2 = 2 elements)
        tdm_load_2d((unsigned)(size_t)(void*)&staps[0],
                    taps + (size_t)b * MAX_TAPS,
                    (unsigned)(2 * count), 1u, (unsigned)(2 * count));
#if __has_builtin(__builtin_amdgcn_s_wait_tensorcnt)
        __builtin_amdgcn_s_wait_tensorcnt(0);
#else
        asm volatile("s_wait_tensorcnt 0x0" ::: "memory");
#endif
    }
#else
    // Fallback: cooperative global->LDS copies
    for (int t = tid; t < count; t += 256)
        staps[t] = taps[(size_t)b * MAX_TAPS + t];
    {
        const int total = ycount * rowElems;
        for (int i = tid; i < total; i += 256) {
            const int r = i / rowElems;
            const int cc = i - r * rowElems;
            tile[i] = src[(size_t)r * (IMG * CH) + cc];
        }
    }
#endif
    __syncthreads();

    // Each thread computes 4 output pixels (32x32 tile / 256 threads).
    for (int p = tid; p < TILE * TILE; p += 256) {
        const int ly = p >> 5;
        const int lx = p & 31;
        const int oy = ty0 + ly;
        const int ox = tx0 + lx;

        float a0 = 0.0f, a1 = 0.0f, a2 = 0.0f;
        for (int t = 0; t < count; ++t) {
            const float2 tp = staps[t];
            const unsigned packed = (unsigned)__float_as_int(tp.y);
            const int dy = (int)(short)(packed >> 16);
            const int dx = (int)(short)(packed & 0xFFFFu);
            const int ry = oy + dy - y0;
            const int rx = ox + dx - x0;
            if ((unsigned)ry < (unsigned)ycount &&
                (unsigned)rx < (unsigned)xcount) {
                const float* __restrict__ pp = &tile[ry * rowElems + rx * CH];
                a0 = fmaf(tp.x, pp[0], a0);
                a1 = fmaf(tp.x, pp[1], a1);
                a2 = fmaf(tp.x, pp[2], a2);
            }
        }
        const size_t o = (((size_t)b * IMG + oy) * IMG + ox) * CH;
        out[o + 0] = a0;
        out[o + 1] = a1;
        out[o + 2] = a2;
    }
}

// ---------------------------------------------------------------------------
extern "C" void kernel_launch(void* const* d_in, const int* in_sizes, int n_in,
                              void* d_out, int out_size, void* d_ws, size_t ws_size,
                              hipStream_t stream) {
    const float* x      = (const float*)d_in[0];   // (128,224,224,3) f32
    const float* table  = (const float*)d_in[1];   // (32,32,32,3) f32
    const int*   amt    = (const int*)d_in[2];     // (128,) i32
    const int*   angles = (const int*)d_in[3];     // (128,) i32
    float* out = (float*)d_out;

    int*    counts = (int*)d_ws;                         // 128 * 4 B
    float2* taps   = (float2*)((char*)d_ws + 512);       // 128 * 1024 * 8 B

    prep_taps_kernel<<<dim3(NB), dim3(32), 0, stream>>>(
        table, amt, angles, counts, taps);

    dim3 grid(IMG / TILE, IMG / TILE, NB);   // 7 x 7 x 128
    blur_conv_kernel<<<grid, dim3(256), 0, stream>>>(x, counts, taps, out);
}